// CRF_4209067950475
// MI455X (gfx1250) — compile-verified
//
#include <hip/hip_runtime.h>
#include <math.h>

#define SEQ   512
#define BATCH 1024
#define T     48
#define RP    50   // padded LDS row stride (floats): 50*M mod 64 distinct for M=0..15
#define PFD   8    // prefetch depth (steps ahead)

#define LOG2E 1.4426950408889634f
#define LN2   0.6931471805599453f

typedef float v2f __attribute__((ext_vector_type(2)));
typedef float v8f __attribute__((ext_vector_type(8)));

// ---------------------------------------------------------------------------
// Normalizer: forward algorithm in exp2-domain via f32 WMMA.
// One wave (32 lanes) owns a 16-row batch tile for all 512 steps.
// Scores kept in log2 domain: nxt2 = m2 + log2(p @ E) + emis*log2e.
// blockDim = 128 (4 waves), gridDim = 16  -> 64 waves total.
// ---------------------------------------------------------------------------
__global__ __launch_bounds__(128) void crf_normalizer(
    const float* __restrict__ emissions,   // [SEQ][BATCH][T]
    const int*   __restrict__ mask,        // [SEQ][BATCH]
    const float* __restrict__ start_t,     // [T]
    const float* __restrict__ end_t,       // [T]
    const float* __restrict__ trans,       // [T][T]
    float*       __restrict__ den_out)     // [BATCH]
{
    __shared__ float Et[T * RP];           // Et[j][i] = exp(trans[i][j])  (B-matrix, transposed)
    __shared__ float s_score[4][16 * RP];  // per-wave score tiles (log2 domain)

    const int tid  = threadIdx.x;
    const int wave = tid >> 5;
    const int lane = tid & 31;
    const int b0   = (blockIdx.x * 4 + wave) * 16;
    const int nn   = lane & 15;            // A-matrix M / B-&-C-matrix N-within-tile
    const int kh   = lane >> 4;            // K-half selector / C-matrix M-half

    // Stage E^T = exp(transitions)^T into LDS (cooperative, read-only afterwards)
    for (int idx = tid; idx < T * T; idx += 128) {
        int i = idx / T, j = idx % T;
        Et[j * RP + i] = __builtin_amdgcn_exp2f(trans[i * T + j] * LOG2E);
    }
    __syncthreads();   // only barrier needed: Et is read-only below; score tiles are wave-private

    float* score = s_score[wave];

    // init (log2 domain): score[M][j] = (start[j] + emis[0][b0+M][j]) * log2e
    for (int idx = lane; idx < 16 * T; idx += 32) {
        int M = idx / T, j = idx % T;
        score[M * RP + j] = (start_t[j] + emissions[(size_t)(b0 + M) * T + j]) * LOG2E;
    }

    // Hoist B-matrix operands into registers: 3 N-tiles x 12 K-blocks.
    // B 4x16 f32 layout: element(K,N): vgpr = K%2, lane = 16*(K/2) + N.
    v2f breg[3][12];
#pragma unroll
    for (int nt = 0; nt < 3; ++nt)
#pragma unroll
        for (int kb = 0; kb < 12; ++kb)
            breg[nt][kb] = *(const v2f*)&Et[(nt * 16 + nn) * RP + kb * 4 + kh * 2];

    // Software pipeline registers: emissions/mask for the step being computed.
    float e_cur[3][8];
    int   mk_cur[8];
#pragma unroll
    for (int v = 0; v < 8; ++v) {
        const int b = b0 + v + 8 * kh;
        mk_cur[v] = mask[1 * BATCH + b];
        const float* ep = emissions + ((size_t)1 * BATCH + b) * T + nn;
#pragma unroll
        for (int nt = 0; nt < 3; ++nt) e_cur[nt][v] = ep[nt * 16];
    }

    for (int t = 1; t < SEQ; ++t) {
        // Deep prefetch: tile (t+PFD) is one contiguous 3072B block -> 24 x 128B lines
        if (t + PFD < SEQ && lane < 24) {
            const char* pf = (const char*)(emissions + ((size_t)(t + PFD) * BATCH + b0) * T)
                             + lane * 128;
            __builtin_prefetch(pf, 0, 1);
        }

        // ---- phase 1: strided score reads -> row max -> exp2 directly into A registers
        // lane covers row nn, columns {4*kb + 2*kh, +1 : kb=0..11}
        v2f sv[12];
#pragma unroll
        for (int kb = 0; kb < 12; ++kb)
            sv[kb] = *(const v2f*)&score[nn * RP + kb * 4 + kh * 2];
        float mx = -INFINITY;
#pragma unroll
        for (int kb = 0; kb < 12; ++kb) mx = fmaxf(mx, fmaxf(sv[kb].x, sv[kb].y));
        mx = fmaxf(mx, __shfl_xor(mx, 16, 32));     // two lanes/row cover complementary cols
        v2f a[12];
#pragma unroll
        for (int kb = 0; kb < 12; ++kb) {
            a[kb].x = __builtin_amdgcn_exp2f(sv[kb].x - mx);
            a[kb].y = __builtin_amdgcn_exp2f(sv[kb].y - mx);
        }

        // ---- issue next step's emissions/mask loads (consumed next iteration)
        const int tn = (t + 1 < SEQ) ? t + 1 : t;
        float e_nxt[3][8];
        int   mk_nxt[8];
#pragma unroll
        for (int v = 0; v < 8; ++v) {
            const int b = b0 + v + 8 * kh;
            mk_nxt[v] = mask[tn * BATCH + b];
            const float* ep = emissions + ((size_t)tn * BATCH + b) * T + nn;
#pragma unroll
            for (int nt = 0; nt < 3; ++nt) e_nxt[nt][v] = ep[nt * 16];
        }

        // ---- phase 2: (16x48 p) @ (48x48 E) via WMMA f32 16x16x4, fused write-back
#pragma unroll
        for (int nt = 0; nt < 3; ++nt) {
            v8f acc = {};
#pragma unroll
            for (int kb = 0; kb < 12; ++kb)
                acc = __builtin_amdgcn_wmma_f32_16x16x4_f32(
                    false, a[kb], false, breg[nt][kb], (short)0, acc, false, false);
            // C/D layout: vgpr v, lanes 0-15: (M=v, N=lane); lanes 16-31: (M=v+8, N=lane-16)
            const int j = nn + nt * 16;
#pragma unroll
            for (int v = 0; v < 8; ++v) {
                const int M = v + 8 * kh;
                const float mrow = __shfl(mx, v + 8 * kh, 32);  // lane r holds row r's max
                const float old  = score[M * RP + j];
                // acc >= exp2(-0.145): raw v_log_f32 is safe (no denorm guard needed)
                const float nv   = e_cur[nt][v] * LOG2E + mrow + __builtin_amdgcn_logf(acc[v]);
                score[M * RP + j] = mk_cur[v] ? nv : old;
            }
        }

        // rotate pipeline
#pragma unroll
        for (int v = 0; v < 8; ++v) {
            mk_cur[v] = mk_nxt[v];
#pragma unroll
            for (int nt = 0; nt < 3; ++nt) e_cur[nt][v] = e_nxt[nt][v];
        }
    }

    // final: den[b] = logsumexp_j(score[b][j] + end[j]); lane's strided 24-col set + partner
    {
        v2f sv[12];
        v2f ev[12];
#pragma unroll
        for (int kb = 0; kb < 12; ++kb) {
            const int c = kb * 4 + kh * 2;
            sv[kb] = *(const v2f*)&score[nn * RP + c];
            ev[kb].x = end_t[c] * LOG2E;
            ev[kb].y = end_t[c + 1] * LOG2E;
        }
        float mx = -INFINITY;
#pragma unroll
        for (int kb = 0; kb < 12; ++kb)
            mx = fmaxf(mx, fmaxf(sv[kb].x + ev[kb].x, sv[kb].y + ev[kb].y));
        mx = fmaxf(mx, __shfl_xor(mx, 16, 32));
        float sum = 0.f;
#pragma unroll
        for (int kb = 0; kb < 12; ++kb) {
            sum += __builtin_amdgcn_exp2f(sv[kb].x + ev[kb].x - mx);
            sum += __builtin_amdgcn_exp2f(sv[kb].y + ev[kb].y - mx);
        }
        sum += __shfl_xor(sum, 16, 32);
        if (kh == 0)
            den_out[b0 + nn] = (mx + __builtin_amdgcn_logf(sum)) * LN2;
    }
}

// ---------------------------------------------------------------------------
// Numerator: per-batch gather score (runs after normalizer; emissions L2-hot)
// ---------------------------------------------------------------------------
__global__ __launch_bounds__(256) void crf_score(
    const float* __restrict__ emissions, const int* __restrict__ tags,
    const int* __restrict__ mask, const float* __restrict__ start_t,
    const float* __restrict__ end_t, const float* __restrict__ trans,
    float* __restrict__ num_out)
{
    const int b = blockIdx.x * blockDim.x + threadIdx.x;
    if (b >= BATCH) return;

    int prev = tags[b];
    float sc = start_t[prev] + emissions[(size_t)b * T + prev];
    int cnt = mask[b] ? 1 : 0;
    for (int t = 1; t < SEQ; ++t) {
        const int tg = tags[t * BATCH + b];
        const int mk = mask[t * BATCH + b];
        if (mk) {
            sc += trans[prev * T + tg] + emissions[((size_t)t * BATCH + b) * T + tg];
            ++cnt;
        }
        prev = tg;
    }
    const int ltag = tags[(cnt - 1) * BATCH + b];
    num_out[b] = sc + end_t[ltag];
}

// ---------------------------------------------------------------------------
// Final reduction: mean(num - den) over BATCH
// ---------------------------------------------------------------------------
__global__ __launch_bounds__(1024) void crf_reduce(
    const float* __restrict__ num, const float* __restrict__ den,
    float* __restrict__ out)
{
    __shared__ float buf[32];
    const int tid = threadIdx.x;
    float v = num[tid] - den[tid];
#pragma unroll
    for (int o = 16; o > 0; o >>= 1) v += __shfl_xor(v, o, 32);
    if ((tid & 31) == 0) buf[tid >> 5] = v;
    __syncthreads();
    if (tid < 32) {
        float w = buf[tid];
#pragma unroll
        for (int o = 16; o > 0; o >>= 1) w += __shfl_xor(w, o, 32);
        if (tid == 0) out[0] = w / (float)BATCH;
    }
}

extern "C" void kernel_launch(void* const* d_in, const int* in_sizes, int n_in,
                              void* d_out, int out_size, void* d_ws, size_t ws_size,
                              hipStream_t stream)
{
    const float* emissions = (const float*)d_in[0];  // [SEQ][BATCH][T] f32
    const int*   tags      = (const int*)  d_in[1];  // [SEQ][BATCH] i32
    const int*   mask      = (const int*)  d_in[2];  // [SEQ][BATCH] i32
    const float* start_t   = (const float*)d_in[3];  // [T]
    const float* end_t     = (const float*)d_in[4];  // [T]
    const float* trans     = (const float*)d_in[5];  // [T][T]

    float* den = (float*)d_ws;          // BATCH floats
    float* num = den + BATCH;           // BATCH floats

    crf_normalizer<<<BATCH / 64, 128, 0, stream>>>(emissions, mask, start_t, end_t, trans, den);
    crf_score<<<BATCH / 256, 256, 0, stream>>>(emissions, tags, mask, start_t, end_t, trans, num);
    crf_reduce<<<1, 1024, 0, stream>>>(num, den, (float*)d_out);
}